// Multi_cat_52458730554099
// MI455X (gfx1250) — compile-verified
//
#include <hip/hip_runtime.h>
#include <math.h>

typedef __attribute__((ext_vector_type(2))) float v2f;
typedef __attribute__((ext_vector_type(4))) float v4f;
typedef __attribute__((ext_vector_type(8))) float v8f;

#define S_SLICES 64

// ---------------------------------------------------------------------------
// Phase 1: banded Gram matrix via V_WMMA_F32_16X16X4_F32 (fp32 in, fp32 acc).
// Each 32-thread wave owns a 16-slice row block (blockIdx.y) and a K-chunk
// (blockIdx.x). A-matrix layout (ISA 7.12.2, 32-bit A 16x4): lane m (0..15)
// holds row s0+m at K={0,1}; lane m+16 holds the SAME row at K={2,3} -> each
// row is streamed contiguously by a lane pair as float2 loads.
// Two WMMAs per step: B=A gives self-dots on the tile diagonal; B=rows+1
// gives adjacent dots on the diagonal. Partials are written per-chunk and
// reduced in fixed order later (deterministic, no float atomics).
// ---------------------------------------------------------------------------
__global__ __launch_bounds__(32)
void band_dots_wmma(const float* __restrict__ q, float* __restrict__ part,
                    int D, int kchunk, int nchunks) {
  const int lane  = threadIdx.x;       // 0..31, full wave, no divergence
  const int m     = lane & 15;         // row within tile
  const int khalf = lane >> 4;         // 0: K0,K1   1: K2,K3
  const int job   = blockIdx.y;        // 0..3 -> rows s0..s0+15
  const int s0    = job * 16;
  const int chunk = blockIdx.x;
  const int k0    = chunk * kchunk;
  int kend = k0 + kchunk;
  if (kend > D) kend = D;

  const int rowA   = s0 + m;           // A row, and B-self column
  const int rowB   = s0 + 1 + m;       // B-adjacent column (slice s+1)
  const float bmsk = (rowB < S_SLICES) ? 1.0f : 0.0f;  // zero dot(63, 64)
  const int rowBs  = (rowB < S_SLICES) ? rowB : rowA;  // safe address

  const float* pA = q + (size_t)rowA  * (size_t)D + 2 * khalf;
  const float* pB = q + (size_t)rowBs * (size_t)D + 2 * khalf;

  v8f accS = {};   // G[s0+m][s0+n]   (self block)
  v8f accA = {};   // G[s0+m][s0+1+n] (shifted block)
#pragma unroll 4
  for (int k = k0; k < kend; k += 4) {
    v2f a  = *(const v2f*)(pA + k);    // rowA, K = k+2*khalf .. +1
    v2f ba = *(const v2f*)(pB + k);
    ba *= bmsk;
    // 8 args: (neg_a, A, neg_b, B, c_mod, C, reuse_a, reuse_b)
    accS = __builtin_amdgcn_wmma_f32_16x16x4_f32(
        false, a, false, a,  (short)0, accS, false, false);
    accA = __builtin_amdgcn_wmma_f32_16x16x4_f32(
        false, a, false, ba, (short)0, accA, false, false);
  }

  // C/D layout: vreg r, lanes 0-15 -> (M=r, N=lane); lanes 16-31 -> (M=8+r,
  // N=lane-16). Diagonal elements live in lanes 0-7 (r=lane) and 24-31
  // (r=lane-24); in both cases r = lane&7 and M = lane&15.
  if (lane < 8 || lane >= 24) {
    const int r = lane & 7;
    union { v8f v; float f[8]; } uS, uA;
    uS.v = accS;
    uA.v = accA;
    const int oS = (job * 2 + 0) * 16 + m;   // self[s0+m] slot
    const int oA = (job * 2 + 1) * 16 + m;   // adj[s0+m]  slot
    part[(size_t)oS * nchunks + chunk] = uS.f[r];
    part[(size_t)oA * nchunks + chunk] = uA.f[r];
  }
}

// ---------------------------------------------------------------------------
// Phase 1b: fixed-order reduction of per-chunk partials -> dots[0..63]=self,
// dots[64..127]=adj (adj[s] = dot(q[s], q[s+1])).
// ---------------------------------------------------------------------------
__global__ __launch_bounds__(128)
void reduce_partials(const float* __restrict__ part, float* __restrict__ dots,
                     int nchunks) {
  const int o = threadIdx.x;           // 0..127
  if (o >= 128) return;
  const float* p = part + (size_t)o * (size_t)nchunks;
  float sum = 0.0f;
  for (int c = 0; c < nchunks; ++c) sum += p[c];
  const int job  = o >> 5;
  const int kind = (o >> 4) & 1;       // 0=self, 1=adj
  const int m    = o & 15;
  dots[kind * 64 + job * 16 + m] = sum;
}

// ---------------------------------------------------------------------------
// Phase 2: entropy branch ladder + 2-way softmax (matches reference tie order
// and boundary overrides). Writes wts[0..63]=a1, wts[64..127]=a2 and kidx.
// ---------------------------------------------------------------------------
__global__ __launch_bounds__(64)
void attn_weights(const float* __restrict__ e, const float* __restrict__ dots,
                  float* __restrict__ wts, int* __restrict__ kidx, int D) {
  const int s = threadIdx.x;
  if (s >= S_SLICES) return;
  int k1, k2;
  if (s == 0) {
    const int b = (e[0] >= e[1]) ? 1 : 0;
    k1 = b; k2 = b;
  } else if (s == S_SLICES - 1) {
    const int b = (e[s] >= e[s - 1]) ? (s - 1) : s;
    k1 = b; k2 = b;
  } else {
    const float ec = e[s], ep = e[s - 1], en = e[s + 1];
    const bool ge_n = ec >= en, ge_p = ec >= ep;
    const bool le_n = ec <= en, le_p = ec <= ep;
    if (ge_n && ge_p)      { k1 = s - 1; k2 = s + 1; }   // local max
    else if (le_n && ge_p) { k1 = s - 1; k2 = s;     }   // rising
    else if (ge_n && le_p) { k1 = s;     k2 = s + 1; }   // falling
    else                   { k1 = s;     k2 = s;     }
  }
  auto G = [&](int t) -> float {       // banded Gram lookup, t in {s-1,s,s+1}
    if (t == s)     return dots[s];
    if (t == s + 1) return dots[64 + s];
    return dots[64 + (s - 1)];         // symmetric: dot(q[s-1], q[s])
  };
  const float scale = 1.0f / sqrtf((float)D);
  const float s1 = G(k1) * scale;
  const float s2 = G(k2) * scale;
  const float mx = fmaxf(s1, s2);
  const float e1 = expf(s1 - mx);
  const float e2 = expf(s2 - mx);
  const float inv = 1.0f / (e1 + e2);
  wts[s]      = e1 * inv;
  wts[64 + s] = e2 * inv;
  kidx[s]      = k1;
  kidx[64 + s] = k2;
}

// ---------------------------------------------------------------------------
// Phase 3: out[s] = a1*q[k1] + a2*q[k2]. float4 streaming; neighbor reads hit
// the 192MB L2 (whole input fits); write-once output uses non-temporal stores
// so it does not evict the input working set.
// ---------------------------------------------------------------------------
__global__ __launch_bounds__(256)
void combine(const float* __restrict__ q, const float* __restrict__ wts,
             const int* __restrict__ kidx, float* __restrict__ out, int D) {
  const int s = blockIdx.y;
  const long long d4 = (long long)blockIdx.x * 256 + threadIdx.x;
  const long long n4 = (long long)D >> 2;
  if (d4 >= n4) return;
  const float a1 = wts[s];
  const float a2 = wts[64 + s];
  const int   k1 = kidx[s];
  const int   k2 = kidx[64 + s];
  const v4f* q1 = (const v4f*)(q + (size_t)k1 * (size_t)D);
  const v4f* q2 = (const v4f*)(q + (size_t)k2 * (size_t)D);
  v4f r = a1 * q1[d4] + a2 * q2[d4];
  __builtin_nontemporal_store(r, (v4f*)(out + (size_t)s * (size_t)D) + d4);
}

// ---------------------------------------------------------------------------
extern "C" void kernel_launch(void* const* d_in, const int* in_sizes, int n_in,
                              void* d_out, int out_size, void* d_ws,
                              size_t ws_size, hipStream_t stream) {
  const float* q = (const float*)d_in[0];   // [S, C, H, W] fp32
  const float* e = (const float*)d_in[1];   // [S] fp32
  float* out = (float*)d_out;

  const int S = in_sizes[1];                // 64
  const int D = in_sizes[0] / S;            // 524288 (multiple of 4)

  const int kchunk  = 2048;
  const int nchunks = (D + kchunk - 1) / kchunk;   // 256 for D=524288

  // Workspace layout (all fp32-sized slots, fully overwritten every call):
  float* part = (float*)d_ws;                        // 128 * nchunks
  float* dots = part + (size_t)128 * (size_t)nchunks; // 128
  float* wts  = dots + 128;                           // 128
  int*   kidx = (int*)(wts + 128);                    // 128

  dim3 g1((unsigned)nchunks, 4);
  band_dots_wmma<<<g1, 32, 0, stream>>>(q, part, D, kchunk, nchunks);
  reduce_partials<<<1, 128, 0, stream>>>(part, dots, nchunks);
  attn_weights<<<1, 64, 0, stream>>>(e, dots, wts, kidx, D);
  dim3 g3((unsigned)((D / 4 + 255) / 256), (unsigned)S);
  combine<<<g3, 256, 0, stream>>>(q, wts, kidx, out, D);
}